// ResidualGNNs2_66975720013955
// MI455X (gfx1250) — compile-verified
//
#include <hip/hip_runtime.h>
#include <math.h>

#define B_    16
#define N_    1000
#define NT    16000
#define E_    1024000
#define HC    64
#define HL    192
#define HID   256
#define TRIU  499500
#define KCHUNK 2048

typedef __attribute__((ext_vector_type(2))) float v2f;
typedef __attribute__((ext_vector_type(8))) float v8f;

// ---------------- degree / norm ----------------
__global__ void k_deg_init(float* deg) {
  int i = blockIdx.x * blockDim.x + threadIdx.x;
  if (i < NT) deg[i] = 1.0f;  // self loop
}
__global__ void k_deg_edges(const int* __restrict__ ei, float* deg) {
  int e = blockIdx.x * blockDim.x + threadIdx.x;
  if (e < E_) atomicAdd(&deg[ei[E_ + e]], 1.0f);
}
__global__ void k_dinv(const float* __restrict__ deg, float* dinv) {
  int i = blockIdx.x * blockDim.x + threadIdx.x;
  if (i < NT) dinv[i] = rsqrtf(deg[i]);  // deg >= 1 always (self loops)
}

// ---------------- generic f32 WMMA GEMM: C[M,N] = A[M,K] @ W[K,N] ----------------
// grid = (M/16, N/16), block = 32 (one wave). K % 4 == 0.
__global__ void gemm_wmma(const float* __restrict__ A, const float* __restrict__ W,
                          float* __restrict__ C, int K, int Ncols) {
  const int mt = blockIdx.x, nt = blockIdx.y;
  const int l = threadIdx.x;
  const int n16 = l & 15;        // M index for A lanes, N index for B/C lanes
  const int half = l >> 4;       // 0: K=k0,k0+1 ; 1: K=k0+2,k0+3
  v8f c = {};
  const float* Arow = A + (size_t)(mt * 16 + n16) * K;
  for (int k0 = 0; k0 < K; k0 += 4) {
    const int ka = k0 + half * 2;
    v2f a; a.x = Arow[ka]; a.y = Arow[ka + 1];
    const float* Wr = W + (size_t)ka * Ncols + nt * 16 + n16;
    v2f b; b.x = Wr[0]; b.y = Wr[Ncols];
    c = __builtin_amdgcn_wmma_f32_16x16x4_f32(false, a, false, b, (short)0, c,
                                              false, false);
  }
#pragma unroll
  for (int i = 0; i < 8; ++i) {
    const int row = mt * 16 + i + half * 8;
    C[(size_t)row * Ncols + nt * 16 + n16] = c[i];
  }
}

// ---------------- GCN message passing ----------------
__global__ void k_selfinit(const float* __restrict__ h, const float* __restrict__ dinv,
                           float* __restrict__ acc) {
  int idx = blockIdx.x * blockDim.x + threadIdx.x;
  if (idx < NT * HC) {
    int i = idx >> 6;
    float d = dinv[i];
    acc[idx] = h[idx] * d * d;
  }
}
__global__ void k_edges(const int* __restrict__ ei, const float* __restrict__ h,
                        const float* __restrict__ dinv, float* __restrict__ acc) {
  long long idx = (long long)blockIdx.x * blockDim.x + threadIdx.x;
  if (idx < (long long)E_ * HC) {
    int e = (int)(idx >> 6);
    int f = (int)(idx & 63);
    int src = ei[e], dst = ei[E_ + e];
    float nrm = dinv[src] * dinv[dst];
    atomicAdd(&acc[(size_t)dst * HC + f], h[(size_t)src * HC + f] * nrm);
  }
}
__global__ void k_gcn_fin(const float* __restrict__ acc, const float* __restrict__ bias,
                          const int* __restrict__ batch, float* __restrict__ hout,
                          float* __restrict__ pooled, int layerOff) {
  int idx = blockIdx.x * blockDim.x + threadIdx.x;
  if (idx < NT * HC) {
    int i = idx >> 6, f = idx & 63;
    float v = tanhf(acc[idx] + bias[f]);
    hout[idx] = v;
    atomicAdd(&pooled[batch[i] * HL + layerOff + f], v);
  }
}

// ---------------- pooled BN + hsum ----------------
__global__ void k_bnh(const float* __restrict__ pooled, const float* __restrict__ g,
                      const float* __restrict__ b, float* __restrict__ hbn) {
  int f = threadIdx.x;
  if (f < HL) {
    float s = 0.f, s2 = 0.f;
    for (int r = 0; r < B_; ++r) { float v = pooled[r * HL + f]; s += v; s2 += v * v; }
    float m = s / B_, var = s2 / B_ - m * m;
    float rstd = rsqrtf(var + 1e-5f);
    for (int r = 0; r < B_; ++r)
      hbn[r * HL + f] = g[f] * (pooled[r * HL + f] - m) * rstd + b[f];
  }
}
__global__ void k_hsum(const float* __restrict__ hbn, const float* __restrict__ mask,
                       float* __restrict__ hsum) {
  int r = threadIdx.x;
  if (r < B_) {
    float s = 0.f;
    for (int f = 0; f < HL; ++f) s += hbn[r * HL + f];
    hsum[r] = mask[r] * s;  // softmax over singleton axis == 1
  }
}

// ---------------- triu offsets + per-column BN stats ----------------
__device__ __forceinline__ int triu_rowstart(int i) { return i * (N_ - 1) - (i * (i - 1)) / 2; }

__global__ void k_triu_off(int* __restrict__ toff) {
  int t = blockIdx.x * blockDim.x + threadIdx.x;
  if (t < TRIU) {
    double D = 2.0 * N_ - 1.0;
    int i = (int)((D - sqrt(D * D - 8.0 * (double)t)) * 0.5);
    if (i < 0) i = 0;
    if (i > N_ - 2) i = N_ - 2;
    while (i > 0 && triu_rowstart(i) > t) --i;
    while (i < N_ - 2 && triu_rowstart(i + 1) <= t) ++i;
    int ju = i + 1 + (t - triu_rowstart(i));
    toff[t] = i * N_ + ju;
  }
}
__global__ void k_xt_stats(const float* __restrict__ x, const int* __restrict__ toff,
                           const float* __restrict__ bn_g, const float* __restrict__ bn_b,
                           float* __restrict__ scaleA, float* __restrict__ shiftA) {
  int t = blockIdx.x * blockDim.x + threadIdx.x;
  if (t < TRIU) {
    int off = toff[t];
    float s = 0.f, s2 = 0.f;
    for (int r = 0; r < B_; ++r) {
      float v = x[(size_t)r * (N_ * N_) + off];
      s += v; s2 += v * v;
    }
    float m = s / B_, var = s2 / B_ - m * m;
    float sc = bn_g[t] * rsqrtf(var + 1e-5f);
    scaleA[t] = sc;
    shiftA[t] = bn_b[t] - m * sc;
  }
}

// ---------------- big split-K WMMA GEMM: [16 x TRIU] (gathered, BN-fused) @ m1_w ----------------
// grid = (numKchunks, HID/16), block = 32
__global__ void mlp1_wmma(const float* __restrict__ x, const int* __restrict__ toff,
                          const float* __restrict__ scaleA, const float* __restrict__ shiftA,
                          const float* __restrict__ W, float* __restrict__ acc) {
  const int kc = blockIdx.x, nt = blockIdx.y;
  const int l = threadIdx.x;
  const int n16 = l & 15;
  const int half = l >> 4;
  int kbeg = kc * KCHUNK;
  int kend = kbeg + KCHUNK;
  if (kend > TRIU) kend = TRIU;
  const size_t rowbase = (size_t)n16 * (N_ * N_);
  v8f c = {};
  for (int k0 = kbeg; k0 < kend; k0 += 4) {
    const int ka = k0 + half * 2;
    int o0 = toff[ka], o1 = toff[ka + 1];
    v2f a;
    a.x = x[rowbase + o0] * scaleA[ka] + shiftA[ka];
    a.y = x[rowbase + o1] * scaleA[ka + 1] + shiftA[ka + 1];
    const float* Wr = W + (size_t)ka * HID + nt * 16 + n16;
    v2f b; b.x = Wr[0]; b.y = Wr[HID];
    c = __builtin_amdgcn_wmma_f32_16x16x4_f32(false, a, false, b, (short)0, c,
                                              false, false);
  }
#pragma unroll
  for (int i = 0; i < 8; ++i) {
    const int row = i + half * 8;
    atomicAdd(&acc[row * HID + nt * 16 + n16], c[i]);
  }
}

// ---------------- MLP finalizers ----------------
__global__ void k_m1_fin(const float* __restrict__ acc, const float* __restrict__ wlast,
                         const float* __restrict__ hsum, const float* __restrict__ bias,
                         const float* __restrict__ g, const float* __restrict__ be,
                         float* __restrict__ z) {
  int n = threadIdx.x;
  if (n < HID) {
    float raw[B_];
    float s = 0.f, s2 = 0.f;
    for (int r = 0; r < B_; ++r) {
      float v = acc[r * HID + n] + hsum[r] * wlast[n] + bias[n];
      raw[r] = v; s += v; s2 += v * v;
    }
    float m = s / B_, var = s2 / B_ - m * m;
    float rstd = rsqrtf(var + 1e-5f);
    for (int r = 0; r < B_; ++r) {
      float v = g[n] * (raw[r] - m) * rstd + be[n];
      z[r * HID + n] = v > 0.f ? v : 0.f;
    }
  }
}
__global__ void k_bn_relu(const float* __restrict__ raw, const float* __restrict__ bias,
                          const float* __restrict__ g, const float* __restrict__ be,
                          float* __restrict__ z, int cols) {
  int n = threadIdx.x;
  if (n < cols) {
    float buf[B_];
    float s = 0.f, s2 = 0.f;
    for (int r = 0; r < B_; ++r) {
      float v = raw[r * cols + n] + bias[n];
      buf[r] = v; s += v; s2 += v * v;
    }
    float m = s / B_, var = s2 / B_ - m * m;
    float rstd = rsqrtf(var + 1e-5f);
    for (int r = 0; r < B_; ++r) {
      float v = g[n] * (buf[r] - m) * rstd + be[n];
      z[r * cols + n] = v > 0.f ? v : 0.f;
    }
  }
}
__global__ void k_final(const float* __restrict__ z3, const float* __restrict__ w,
                        const float* __restrict__ b, float* __restrict__ out) {
  int t = threadIdx.x;
  if (t < B_ * 2) {
    int r = t >> 1, n = t & 1;
    float s = b[n];
    for (int k = 0; k < 128; ++k) s += z3[r * 128 + k] * w[k * 2 + n];
    out[r * 2 + n] = s;
  }
}

extern "C" void kernel_launch(void* const* d_in, const int* in_sizes, int n_in,
                              void* d_out, int out_size, void* d_ws, size_t ws_size,
                              hipStream_t stream) {
  const float* x     = (const float*)d_in[0];
  const int*   ei    = (const int*)d_in[1];
  const int*   batch = (const int*)d_in[2];
  const float* mask  = (const float*)d_in[3];
  const float* W0 = (const float*)d_in[4];  const float* b0 = (const float*)d_in[5];
  const float* W1 = (const float*)d_in[6];  const float* b1 = (const float*)d_in[7];
  const float* W2 = (const float*)d_in[8];  const float* b2 = (const float*)d_in[9];
  const float* bn_g  = (const float*)d_in[10]; const float* bn_b  = (const float*)d_in[11];
  const float* bnh_g = (const float*)d_in[12]; const float* bnh_b = (const float*)d_in[13];
  // d_in[14]=att_w, d_in[15]=att_b : softmax over singleton axis -> unused
  const float* m1_w = (const float*)d_in[16]; const float* m1_b = (const float*)d_in[17];
  const float* m1_g = (const float*)d_in[18]; const float* m1_be = (const float*)d_in[19];
  const float* m2_w = (const float*)d_in[20]; const float* m2_b = (const float*)d_in[21];
  const float* m2_g = (const float*)d_in[22]; const float* m2_be = (const float*)d_in[23];
  const float* m3_w = (const float*)d_in[24]; const float* m3_b = (const float*)d_in[25];
  const float* m3_g = (const float*)d_in[26]; const float* m3_be = (const float*)d_in[27];
  const float* m4_w = (const float*)d_in[28]; const float* m4_b = (const float*)d_in[29];
  float* out = (float*)d_out;

  // workspace carve-up (floats)
  float* w = (float*)d_ws;
  size_t o = 0;
  float* deg    = w + o; o += NT;
  float* dinv   = w + o; o += NT;
  float* htmp   = w + o; o += (size_t)NT * HC;
  float* acc    = w + o; o += (size_t)NT * HC;
  float* hA     = w + o; o += (size_t)NT * HC;
  float* hB     = w + o; o += (size_t)NT * HC;
  float* pooled = w + o; o += B_ * HL;
  float* hbn    = w + o; o += B_ * HL;
  float* hsum   = w + o; o += B_;
  int*   toff   = (int*)(w + o); o += TRIU;
  float* scaleA = w + o; o += TRIU;
  float* shiftA = w + o; o += TRIU;
  float* m1acc  = w + o; o += B_ * HID;
  float* z1     = w + o; o += B_ * HID;
  float* z2r    = w + o; o += B_ * 128;
  float* z2     = w + o; o += B_ * 128;
  float* z3r    = w + o; o += B_ * 128;
  float* z3     = w + o; o += B_ * 128;

  const int T = 256;
  // degree / norm
  k_deg_init<<<(NT + T - 1) / T, T, 0, stream>>>(deg);
  k_deg_edges<<<(E_ + T - 1) / T, T, 0, stream>>>(ei, deg);
  k_dinv<<<(NT + T - 1) / T, T, 0, stream>>>(deg, dinv);

  hipMemsetAsync(pooled, 0, B_ * HL * sizeof(float), stream);
  hipMemsetAsync(m1acc, 0, B_ * HID * sizeof(float), stream);

  const int NE = (NT * HC + T - 1) / T;
  const int EE = (int)(((long long)E_ * HC + T - 1) / T);

  // GCN layer 1
  gemm_wmma<<<dim3(NT / 16, HC / 16), 32, 0, stream>>>(x, W0, htmp, 1000, HC);
  k_selfinit<<<NE, T, 0, stream>>>(htmp, dinv, acc);
  k_edges<<<EE, T, 0, stream>>>(ei, htmp, dinv, acc);
  k_gcn_fin<<<NE, T, 0, stream>>>(acc, b0, batch, hA, pooled, 0);
  // GCN layer 2
  gemm_wmma<<<dim3(NT / 16, HC / 16), 32, 0, stream>>>(hA, W1, htmp, HC, HC);
  k_selfinit<<<NE, T, 0, stream>>>(htmp, dinv, acc);
  k_edges<<<EE, T, 0, stream>>>(ei, htmp, dinv, acc);
  k_gcn_fin<<<NE, T, 0, stream>>>(acc, b1, batch, hB, pooled, HC);
  // GCN layer 3
  gemm_wmma<<<dim3(NT / 16, HC / 16), 32, 0, stream>>>(hB, W2, htmp, HC, HC);
  k_selfinit<<<NE, T, 0, stream>>>(htmp, dinv, acc);
  k_edges<<<EE, T, 0, stream>>>(ei, htmp, dinv, acc);
  k_gcn_fin<<<NE, T, 0, stream>>>(acc, b2, batch, hA, pooled, 2 * HC);

  // pooled BN + hsum
  k_bnh<<<1, HL, 0, stream>>>(pooled, bnh_g, bnh_b, hbn);
  k_hsum<<<1, B_, 0, stream>>>(hbn, mask, hsum);

  // triu offsets + per-column BN stats
  k_triu_off<<<(TRIU + T - 1) / T, T, 0, stream>>>(toff);
  k_xt_stats<<<(TRIU + T - 1) / T, T, 0, stream>>>(x, toff, bn_g, bn_b, scaleA, shiftA);

  // big memory-bound GEMM: split-K WMMA streaming all 512 MB of m1_w
  const int nkc = (TRIU + KCHUNK - 1) / KCHUNK;  // 244
  mlp1_wmma<<<dim3(nkc, HID / 16), 32, 0, stream>>>(x, toff, scaleA, shiftA, m1_w, m1acc);
  k_m1_fin<<<1, HID, 0, stream>>>(m1acc, m1_w + (size_t)TRIU * HID, hsum, m1_b,
                                  m1_g, m1_be, z1);

  // MLP layers 2-3 (WMMA) + final tiny layer
  gemm_wmma<<<dim3(1, 128 / 16), 32, 0, stream>>>(z1, m2_w, z2r, HID, 128);
  k_bn_relu<<<1, 128, 0, stream>>>(z2r, m2_b, m2_g, m2_be, z2, 128);
  gemm_wmma<<<dim3(1, 128 / 16), 32, 0, stream>>>(z2, m3_w, z3r, 128, 128);
  k_bn_relu<<<1, 128, 0, stream>>>(z3r, m3_b, m3_g, m3_be, z3, 128);
  k_final<<<1, 32, 0, stream>>>(z3, m4_w, m4_b, out);
}